// SSM_8512625181098
// MI455X (gfx1250) — compile-verified
//
#include <hip/hip_runtime.h>
#include <math.h>

typedef float v2f __attribute__((ext_vector_type(2)));
typedef float v8f __attribute__((ext_vector_type(8)));

#define D_MODEL 512
#define D_STATE 64
#define SEQ_L   4096
#define BATCH   16
#define CHUNK   64
#define NCHUNK  (SEQ_L / CHUNK)   // 64
#define CLAMPV  1000000.0f

// ---------------------------------------------------------------------------
// Kernel 1: powers of M = A^T.  W[k] = M^k for k = 0..CHUNK (row-major 64x64).
// One workgroup, trivial cost (64 x 64^3 MACs).
// ---------------------------------------------------------------------------
__global__ void ssm_powers(const float* __restrict__ A, float* __restrict__ W) {
    __shared__ float M1[64 * 64];
    __shared__ float Wp[64 * 64];
    int tid = threadIdx.x;  // 256
    for (int idx = tid; idx < 4096; idx += 256) {
        int i = idx >> 6, j = idx & 63;
        M1[idx] = A[j * 64 + i];               // M = A^T
        float id = (i == j) ? 1.f : 0.f;
        Wp[idx] = id;
        W[idx]  = id;                          // W_0 = I
    }
    __syncthreads();
    for (int k = 1; k <= CHUNK; ++k) {
        float res[16];
        for (int r = 0; r < 16; ++r) {
            int idx = tid + r * 256;
            int i = idx >> 6, j = idx & 63;
            float acc = 0.f;
            for (int t = 0; t < 64; ++t) acc += Wp[i * 64 + t] * M1[t * 64 + j];
            res[r] = acc;
        }
        __syncthreads();
        for (int r = 0; r < 16; ++r) {
            int idx = tid + r * 256;
            Wp[idx] = res[r];
            W[(size_t)k * 4096 + idx] = res[r];
        }
        __syncthreads();
    }
}

// ---------------------------------------------------------------------------
// Kernel 2: u = x @ Bm^T   ([BL,512] x [512,64] -> [BL,64]) via WMMA f32.
// Block = 128 threads (4 waves); each wave owns one 16-col N-tile; K = 512.
// x tile (16x512) staged in LDS with coalesced loads; Bm stays L2-resident.
// ---------------------------------------------------------------------------
__global__ void ssm_proj(const float* __restrict__ x, const float* __restrict__ Bm,
                         float* __restrict__ u) {
    __shared__ float xs[16][D_MODEL];          // 32 KB
    int tid = threadIdx.x;                     // 128
    long rowbase = (long)blockIdx.x * 16;
    for (int idx = tid; idx < 16 * D_MODEL; idx += 128) {
        int r = idx >> 9, d = idx & 511;
        xs[r][d] = x[(rowbase + r) * D_MODEL + d];
    }
    __syncthreads();
    int wave = tid >> 5;
    int lane = tid & 31;
    int hi   = lane >> 4;                      // 0/1
    int am   = lane & 15;                      // A row / B,C col-in-tile
    int ak   = hi << 1;                        // K offset 0 or 2
    int n    = wave * 16 + am;                 // output state index
    const float* brow = Bm + n * D_MODEL;      // Bm[n][*]
    v8f acc = {};
    for (int k = 0; k < D_MODEL; k += 4) {
        v2f a; a[0] = xs[am][k + ak];  a[1] = xs[am][k + ak + 1];
        v2f b; b[0] = brow[k + ak];    b[1] = brow[k + ak + 1];
        acc = __builtin_amdgcn_wmma_f32_16x16x4_f32(false, a, false, b,
                                                    (short)0, acc, false, false);
    }
    for (int v = 0; v < 8; ++v) {
        int m = v + hi * 8;                    // C/D fragment row
        u[(rowbase + m) * D_STATE + n] = acc[v];
    }
}

// ---------------------------------------------------------------------------
// Kernel 3: chunk-local scan, zero entering state.  1024 independent blocks
// (batch x chunk), 64 threads each (thread = one state element).
// ---------------------------------------------------------------------------
__global__ void ssm_chunkscan(const float* __restrict__ u, const float* __restrict__ W,
                              float* __restrict__ slocal, float* __restrict__ F) {
    __shared__ float M1[64 * 64];
    __shared__ float us[CHUNK][64];
    __shared__ float sbuf[64];
    int j = threadIdx.x;                       // 64
    int b = blockIdx.x >> 6;
    int c = blockIdx.x & 63;
    const float* W1 = W + 4096;                // M^1
    for (int idx = j; idx < 4096; idx += 64) {
        M1[idx] = W1[idx];
        int t = idx >> 6;
        us[t][idx & 63] = u[((size_t)(b * SEQ_L + c * CHUNK + t)) * 64 + (idx & 63)];
    }
    float s = 0.f;
    __syncthreads();
    for (int t = 0; t < CHUNK; ++t) {
        sbuf[j] = s;
        __syncthreads();
        float acc = us[t][j];
        for (int i = 0; i < 64; ++i) acc += sbuf[i] * M1[i * 64 + j];
        s = acc;
        slocal[((size_t)(b * SEQ_L + c * CHUNK + t)) * 64 + j] = s;
        __syncthreads();
    }
    F[(b * NCHUNK + c) * 64 + j] = s;
}

// ---------------------------------------------------------------------------
// Kernel 4: carry scan across chunk boundaries.  E_c = entering state of
// chunk c:  E_0 = 0,  E_{c+1} = E_c * M^CHUNK + F_c.   16 blocks x 64 thr.
// ---------------------------------------------------------------------------
__global__ void ssm_carry(const float* __restrict__ F, const float* __restrict__ W,
                          float* __restrict__ E) {
    __shared__ float WT[64 * 64];
    __shared__ float ebuf[64];
    int j = threadIdx.x;                       // 64
    int b = blockIdx.x;
    const float* WTg = W + (size_t)CHUNK * 4096;
    for (int idx = j; idx < 4096; idx += 64) WT[idx] = WTg[idx];
    float e = 0.f;
    __syncthreads();
    for (int c = 0; c < NCHUNK; ++c) {
        E[(b * NCHUNK + c) * 64 + j] = e;
        ebuf[j] = e;
        __syncthreads();
        float acc = F[(b * NCHUNK + c) * 64 + j];
        for (int i = 0; i < 64; ++i) acc += ebuf[i] * WT[i * 64 + j];
        e = acc;
        __syncthreads();
    }
}

// ---------------------------------------------------------------------------
// Kernel 5: correction + output GEMM + GELU + LayerNorm + nan_to_num, fused.
// Block = 256 threads (8 waves) handles 16 rows x D_MODEL.  Each wave owns
// four 16-col N-tiles; K = 64 via WMMA f32.  y never touches HBM.
// ---------------------------------------------------------------------------
__global__ void ssm_out(const float* __restrict__ Cmat, const float* __restrict__ gamma,
                        const float* __restrict__ beta, const float* __restrict__ slocal,
                        const float* __restrict__ E, const float* __restrict__ W,
                        float* __restrict__ out) {
    __shared__ float ss[16][D_STATE];          // corrected states, 4 KB
    __shared__ float ys[16][D_MODEL];          // gelu(y), 32 KB
    __shared__ float Ec[64];
    __shared__ float red_s[16][16];
    __shared__ float red_q[16][16];
    __shared__ float mu_s[16], rs_s[16];
    int tid = threadIdx.x;                     // 256
    long rowbase = (long)blockIdx.x * 16;
    int b   = blockIdx.x >> 8;                 // 256 tiles per batch
    int l0  = (blockIdx.x & 255) * 16;
    int c   = l0 >> 6;
    int tl0 = l0 & 63;                         // 0,16,32,48
    if (tid < 64) Ec[tid] = E[(b * NCHUNK + c) * 64 + tid];
    __syncthreads();

    // corrected states: s_t = slocal_t + E_c * M^(tlocal+1)
    {
        int r  = tid >> 4;                     // 0..15
        int j0 = (tid & 15) * 4;
        const float* Wp = W + (size_t)(tl0 + r + 1) * 4096;
        const float* sl = slocal + (rowbase + r) * 64;
        float v0 = sl[j0], v1 = sl[j0 + 1], v2 = sl[j0 + 2], v3 = sl[j0 + 3];
        for (int i = 0; i < 64; ++i) {
            float e = Ec[i];
            const float* wr = Wp + i * 64 + j0;
            v0 += e * wr[0]; v1 += e * wr[1]; v2 += e * wr[2]; v3 += e * wr[3];
        }
        ss[r][j0] = v0; ss[r][j0 + 1] = v1; ss[r][j0 + 2] = v2; ss[r][j0 + 3] = v3;
    }
    __syncthreads();

    // y = ss @ C^T  (WMMA), fused GELU into LDS
    int wave = tid >> 5, lane = tid & 31, hi = lane >> 4;
    int am = lane & 15, ak = hi << 1;
    int n0 = wave * 64;
    v8f acc0 = {}, acc1 = {}, acc2 = {}, acc3 = {};
    const float* c0 = Cmat + (n0 +  0 + am) * 64;
    const float* c1 = Cmat + (n0 + 16 + am) * 64;
    const float* c2 = Cmat + (n0 + 32 + am) * 64;
    const float* c3 = Cmat + (n0 + 48 + am) * 64;
    for (int k = 0; k < 64; k += 4) {
        v2f a;  a[0]  = ss[am][k + ak]; a[1]  = ss[am][k + ak + 1];
        v2f b0; b0[0] = c0[k + ak];     b0[1] = c0[k + ak + 1];
        v2f b1; b1[0] = c1[k + ak];     b1[1] = c1[k + ak + 1];
        v2f b2; b2[0] = c2[k + ak];     b2[1] = c2[k + ak + 1];
        v2f b3; b3[0] = c3[k + ak];     b3[1] = c3[k + ak + 1];
        acc0 = __builtin_amdgcn_wmma_f32_16x16x4_f32(false, a, false, b0, (short)0, acc0, false, false);
        acc1 = __builtin_amdgcn_wmma_f32_16x16x4_f32(false, a, false, b1, (short)0, acc1, false, false);
        acc2 = __builtin_amdgcn_wmma_f32_16x16x4_f32(false, a, false, b2, (short)0, acc2, false, false);
        acc3 = __builtin_amdgcn_wmma_f32_16x16x4_f32(false, a, false, b3, (short)0, acc3, false, false);
    }
    const float kInvSqrt2 = 0.70710678118654752f;
    for (int v = 0; v < 8; ++v) {
        int m = v + hi * 8;
        float y;
        y = acc0[v]; ys[m][n0 +  0 + am] = 0.5f * y * (1.f + erff(y * kInvSqrt2));
        y = acc1[v]; ys[m][n0 + 16 + am] = 0.5f * y * (1.f + erff(y * kInvSqrt2));
        y = acc2[v]; ys[m][n0 + 32 + am] = 0.5f * y * (1.f + erff(y * kInvSqrt2));
        y = acc3[v]; ys[m][n0 + 48 + am] = 0.5f * y * (1.f + erff(y * kInvSqrt2));
    }
    __syncthreads();

    // LayerNorm over D per row
    {
        int r = tid >> 4, seg = tid & 15;
        float s = 0.f, q = 0.f;
        for (int d = seg * 32; d < seg * 32 + 32; ++d) {
            float yv = ys[r][d];
            s += yv; q += yv * yv;
        }
        red_s[r][seg] = s; red_q[r][seg] = q;
    }
    __syncthreads();
    if (tid < 16) {
        float s = 0.f, q = 0.f;
        for (int k = 0; k < 16; ++k) { s += red_s[tid][k]; q += red_q[tid][k]; }
        float mu  = s * (1.f / 512.f);
        float var = q * (1.f / 512.f) - mu * mu;
        mu_s[tid] = mu;
        rs_s[tid] = rsqrtf(var + 1e-5f);
    }
    __syncthreads();
    for (int idx = tid; idx < 16 * D_MODEL; idx += 256) {
        int r = idx >> 9, d = idx & 511;
        float v = (ys[r][d] - mu_s[r]) * rs_s[r] * gamma[d] + beta[d];
        if (isnan(v)) v = 0.f;
        else if (isinf(v)) v = (v > 0.f) ? CLAMPV : -CLAMPV;
        out[(rowbase + r) * D_MODEL + d] = v;
    }
}

// ---------------------------------------------------------------------------
extern "C" void kernel_launch(void* const* d_in, const int* in_sizes, int n_in,
                              void* d_out, int out_size, void* d_ws, size_t ws_size,
                              hipStream_t stream) {
    const float* x     = (const float*)d_in[0];
    const float* A     = (const float*)d_in[1];
    const float* Bm    = (const float*)d_in[2];
    const float* Cm    = (const float*)d_in[3];
    const float* gamma = (const float*)d_in[4];
    const float* beta  = (const float*)d_in[5];
    float* out = (float*)d_out;

    float* ws     = (float*)d_ws;
    float* W      = ws;                                   // (CHUNK+1)*64*64
    float* u      = W + (size_t)(CHUNK + 1) * 4096;       // BATCH*SEQ_L*64
    float* slocal = u + (size_t)BATCH * SEQ_L * 64;       // BATCH*SEQ_L*64
    float* F      = slocal + (size_t)BATCH * SEQ_L * 64;  // BATCH*NCHUNK*64
    float* E      = F + (size_t)BATCH * NCHUNK * 64;      // BATCH*NCHUNK*64

    ssm_powers   <<<1,                     256, 0, stream>>>(A, W);
    ssm_proj     <<<BATCH * SEQ_L / 16,    128, 0, stream>>>(x, Bm, u);
    ssm_chunkscan<<<BATCH * NCHUNK,         64, 0, stream>>>(u, W, slocal, F);
    ssm_carry    <<<BATCH,                  64, 0, stream>>>(F, W, E);
    ssm_out      <<<BATCH * SEQ_L / 16,    256, 0, stream>>>(Cm, gamma, beta, slocal, E, W, out);
}